// MultiResolutionPointGenerator_12807592477406
// MI455X (gfx1250) — compile-verified
//
#include <hip/hip_runtime.h>
#include <stdint.h>

#define B 4
#define N 8192
#define F 128
#define NP1 4096   // scale 0.5
#define NP2 2048   // scale 0.25

// Flat float offsets into d_out (reference tuple order)
#define O_COORD   0LL
#define O_FEAT    98304LL      // + B*N*3
#define O_OFF     4292608LL    // + B*N*F
#define O_SC05    4292612LL    // + B
#define O_SF05    4341764LL    // + B*NP1*3
#define O_SO05    6438916LL    // + B*NP1*F
#define O_SC025   6438920LL    // + B
#define O_SF025   6463496LL    // + B*NP2*3
#define O_SO025   7512072LL    // + B*NP2*F   (total 7512076)

// ---------------------------------------------------------------------------
// Kernel 1: passthrough copy (coord, feat) + all three offset vectors
// ---------------------------------------------------------------------------
__global__ void copy_passthrough(const float4* __restrict__ coord4,
                                 const float4* __restrict__ feat4,
                                 const int* __restrict__ off_in,
                                 float* __restrict__ out) {
  const long long nc4 = (long long)B * N * 3 / 4;   // 24576
  const long long nf4 = (long long)B * N * F / 4;   // 1048576
  float4* outc = (float4*)(out + O_COORD);
  float4* outf = (float4*)(out + O_FEAT);
  long long stride = (long long)gridDim.x * blockDim.x;
  for (long long i = (long long)blockIdx.x * blockDim.x + threadIdx.x;
       i < nc4 + nf4; i += stride) {
    if (i < nc4) outc[i] = coord4[i];
    else         outf[i - nc4] = feat4[i - nc4];
  }
  if (blockIdx.x == 0 && threadIdx.x < B) {
    int t = threadIdx.x;
    out[O_OFF   + t] = (float)off_in[t];
    out[O_SO05  + t] = (float)((t + 1) * NP1);
    out[O_SO025 + t] = (float)((t + 1) * NP2);
  }
}

// ---------------------------------------------------------------------------
// Kernel 2: farthest-point sampling, one workgroup (32 wave32s) per cloud.
// Coordinates + running min-distance live entirely in VGPRs (8 pts/thread).
// Argmax reduction carries the winning point's coords through the shuffles,
// so no bulk coord array is needed in LDS and only 2 barriers/iteration.
// ---------------------------------------------------------------------------
#define TPB 1024
#define PPT 8

__device__ __forceinline__ void warp_argmax(float& bd, int& bi,
                                            float& bx, float& by, float& bz) {
#pragma unroll
  for (int s = 16; s > 0; s >>= 1) {
    float od = __shfl_xor(bd, s, 32);
    int   oi = __shfl_xor(bi, s, 32);
    float ox = __shfl_xor(bx, s, 32);
    float oy = __shfl_xor(by, s, 32);
    float oz = __shfl_xor(bz, s, 32);
    // jnp.argmax tie-break: prefer the smaller index on equal distance
    bool t = (od > bd) || (od == bd && oi < bi);
    bd = t ? od : bd; bi = t ? oi : bi;
    bx = t ? ox : bx; by = t ? oy : by; bz = t ? oz : bz;
  }
}

__global__ __launch_bounds__(TPB) void fps_kernel(const float* __restrict__ coord,
                                                  int* __restrict__ idx_ws,
                                                  float* __restrict__ out) {
  __shared__ float rd[32], rx[32], ry[32], rz[32];
  __shared__ int   ri[32];
  __shared__ float bcx, bcy, bcz;

  const int b    = blockIdx.x;
  const int tid  = threadIdx.x;
  const int lane = tid & 31;
  const int wave = tid >> 5;

  const float* cb   = coord + (long long)b * N * 3;
  float* sc05  = out + O_SC05  + (long long)b * NP1 * 3;
  float* sc025 = out + O_SC025 + (long long)b * NP2 * 3;
  int*   iw    = idx_ws + b * NP1;

  float px[PPT], py[PPT], pz[PPT], dist[PPT];
#pragma unroll
  for (int k = 0; k < PPT; ++k) {
    int p = (k << 10) + tid;           // strided ownership: idx = k*1024 + tid
    px[k] = cb[3 * p + 0];
    py[k] = cb[3 * p + 1];
    pz[k] = cb[3 * p + 2];
    dist[k] = 1e10f;
  }

  // centroid 0 is always point 0 (owned by tid 0, k==0)
  if (tid == 0) {
    bcx = px[0]; bcy = py[0]; bcz = pz[0];
    iw[0] = 0;
    sc05[0]  = px[0]; sc05[1]  = py[0]; sc05[2]  = pz[0];
    sc025[0] = px[0]; sc025[1] = py[0]; sc025[2] = pz[0];
  }
  __syncthreads();
  float cx = bcx, cy = bcy, cz = bcz;

  for (int j = 1; j < NP1; ++j) {
    float bd = -1.0f; int bi = 0;
    float bx = px[0], by = py[0], bz = pz[0];
#pragma unroll
    for (int k = 0; k < PPT; ++k) {
      float dx = px[k] - cx, dy = py[k] - cy, dz = pz[k] - cz;
      float dd = fmaf(dz, dz, fmaf(dy, dy, dx * dx));
      float nd = fminf(dist[k], dd);
      dist[k] = nd;
      bool t = nd > bd;                    // ascending k keeps smallest index
      bd = t ? nd : bd; bi = t ? ((k << 10) + tid) : bi;
      bx = t ? px[k] : bx; by = t ? py[k] : by; bz = t ? pz[k] : bz;
    }

    warp_argmax(bd, bi, bx, by, bz);
    if (lane == 0) { rd[wave] = bd; ri[wave] = bi;
                     rx[wave] = bx; ry[wave] = by; rz[wave] = bz; }
    __syncthreads();

    if (wave == 0) {                       // 32 waves -> exactly one wave32
      bd = rd[lane]; bi = ri[lane];
      bx = rx[lane]; by = ry[lane]; bz = rz[lane];
      warp_argmax(bd, bi, bx, by, bz);
      if (lane == 0) {
        bcx = bx; bcy = by; bcz = bz;
        iw[j] = bi;
        sc05[3 * j + 0] = bx; sc05[3 * j + 1] = by; sc05[3 * j + 2] = bz;
        if (j < NP2) {
          sc025[3 * j + 0] = bx; sc025[3 * j + 1] = by; sc025[3 * j + 2] = bz;
        }
      }
    }
    __syncthreads();
    cx = bcx; cy = bcy; cz = bcz;
  }
}

// ---------------------------------------------------------------------------
// Kernel 3: feature gather, one 512B row per wave via CDNA5 async DMA:
// global_load_async_to_lds_b128 -> s_wait_asynccnt 0 -> store_async_from_lds.
// Per-wave ASYNCcnt + private LDS tile => no barriers needed.
// ---------------------------------------------------------------------------
#define GROWS_TOTAL (B * NP1 + B * NP2)    // 16384 + 8192 = 24576

__global__ __launch_bounds__(256) void gather_feat(const float* __restrict__ feat,
                                                   const int* __restrict__ idx_ws,
                                                   float* __restrict__ out) {
  __shared__ __align__(16) float tile[8 * F];
  const int tid  = threadIdx.x;
  const int wave = tid >> 5;
  const int lane = tid & 31;
  const int r = blockIdx.x * 8 + wave;     // grid is sized exactly

  int cloud, j;
  float* dstrow;
  if (r < B * NP1) {
    cloud = r >> 12; j = r & (NP1 - 1);
    dstrow = out + O_SF05 + (long long)r * F;
  } else {
    int r2 = r - B * NP1;
    cloud = r2 >> 11; j = r2 & (NP2 - 1);
    dstrow = out + O_SF025 + (long long)r2 * F;
  }
  const int idx = idx_ws[cloud * NP1 + j];
  const float* src = feat + ((long long)cloud * N + idx) * F + lane * 4;
  float* dst = dstrow + lane * 4;

  unsigned long long gsrc = (unsigned long long)(uintptr_t)src;
  unsigned long long gdst = (unsigned long long)(uintptr_t)dst;
  unsigned lds = (unsigned)(uintptr_t)&tile[wave * F + lane * 4];

  asm volatile("global_load_async_to_lds_b128 %0, %1, off"
               :: "v"(lds), "v"(gsrc) : "memory");
  asm volatile("s_wait_asynccnt 0" ::: "memory");
  asm volatile("global_store_async_from_lds_b128 %0, %1, off"
               :: "v"(gdst), "v"(lds) : "memory");
  // S_ENDPGM performs an implicit wait-idle, draining the async store.
}

// ---------------------------------------------------------------------------
extern "C" void kernel_launch(void* const* d_in, const int* in_sizes, int n_in,
                              void* d_out, int out_size, void* d_ws, size_t ws_size,
                              hipStream_t stream) {
  const float* coord = (const float*)d_in[0];
  const float* feat  = (const float*)d_in[1];
  const int*   off   = (const int*)d_in[2];
  float* out    = (float*)d_out;
  int*   idx_ws = (int*)d_ws;              // B*NP1 ints = 64 KB

  copy_passthrough<<<2048, 256, 0, stream>>>((const float4*)coord,
                                             (const float4*)feat, off, out);
  fps_kernel<<<B, TPB, 0, stream>>>(coord, idx_ws, out);
  gather_feat<<<GROWS_TOTAL / 8, 256, 0, stream>>>(feat, idx_ws, out);
}